// CausalDiscoveryLayer_83064667505146
// MI455X (gfx1250) — compile-verified
//
#include <hip/hip_runtime.h>

typedef float v2f __attribute__((ext_vector_type(2)));
typedef float v8f __attribute__((ext_vector_type(8)));

#define N_NODES 27
#define MPAD    32        // M padded to two 16-row WMMA tiles
#define DK      5120      // E (inner GEMM dim)
#define E3      15360     // 3*E

__device__ __forceinline__ v2f ld2(const float* p) { return *(const v2f*)p; }

#define WMMA_F32(C, A, B) \
    C = __builtin_amdgcn_wmma_f32_16x16x4_f32(false, A, false, B, (short)0, C, false, false)

// ---------------------------------------------------------------------------
// Split-K GEMM partial: P[s, 27, Nout] = Xpad[32,Kc] @ W[Nout,Kc]^T (chunk kb)
// One wave = 32(M) x 32(N) tile over one K chunk, V_WMMA_F32_16X16X4_F32.
// Software-pipelined one group ahead: next K-step's loads issue before the
// current WMMAs, so loads overlap the full WMMA block of the previous step.
// ---------------------------------------------------------------------------
__global__ __launch_bounds__(128)
void gemm_xwt_splitk(const float* __restrict__ Xpad,   // [32, K] zero-padded
                     const float* __restrict__ W,      // [Nout, K]
                     float* __restrict__ P,            // [S, 27, Nout] partials
                     int K, int Kc, int Nout)
{
    const int lane = threadIdx.x & 31;
    const int wave = threadIdx.x >> 5;
    const int n0   = (blockIdx.x * 4 + wave) * 32;
    const int kb   = blockIdx.y * Kc;

    const int lHi  = lane >> 4;                  // selects K sub-pair (0/2)
    const int lLo  = lane & 15;
    const int koff = kb + 2 * lHi;

    const float* a0p = Xpad + (size_t)lLo * K + koff;         // rows 0..15
    const float* a1p = Xpad + (size_t)(16 + lLo) * K + koff;  // rows 16..31 (zero-pad)
    const float* b0p = W + (size_t)(n0 + lLo) * K + koff;
    const float* b1p = W + (size_t)(n0 + 16 + lLo) * K + koff;

    v8f c00 = {}; v8f c01 = {}; v8f c10 = {}; v8f c11 = {};

    // prologue loads
    v2f a0 = ld2(a0p), a1 = ld2(a1p), b0 = ld2(b0p), b1 = ld2(b1p);

#pragma unroll 8
    for (int k = 4; k < Kc; k += 4) {
        // issue next group's loads first (independent of current WMMAs)
        v2f na0 = ld2(a0p + k);
        v2f na1 = ld2(a1p + k);
        v2f nb0 = ld2(b0p + k);
        v2f nb1 = ld2(b1p + k);
        WMMA_F32(c00, a0, b0);
        WMMA_F32(c01, a0, b1);
        WMMA_F32(c10, a1, b0);
        WMMA_F32(c11, a1, b1);
        a0 = na0; a1 = na1; b0 = nb0; b1 = nb1;
    }
    // epilogue
    WMMA_F32(c00, a0, b0);
    WMMA_F32(c01, a0, b1);
    WMMA_F32(c10, a1, b0);
    WMMA_F32(c11, a1, b1);

    float* Pp = P + (size_t)blockIdx.y * N_NODES * Nout;
    const int rbase = lHi << 3;                  // C layout: row = vg + 8*(lane/16)
#pragma unroll
    for (int vg = 0; vg < 8; ++vg) {
        const int r0 = vg + rbase;               // 0..15
        const int r1 = r0 + 16;                  // 16..31
        Pp[(size_t)r0 * Nout + n0 + lLo]      = c00[vg];
        Pp[(size_t)r0 * Nout + n0 + 16 + lLo] = c01[vg];
        if (r1 < N_NODES) {
            Pp[(size_t)r1 * Nout + n0 + lLo]      = c10[vg];
            Pp[(size_t)r1 * Nout + n0 + 16 + lLo] = c11[vg];
        }
    }
}

// ---------------------------------------------------------------------------
// Y[m,n] = (sum_s P[s,m,n] + bias[n]) * scale   (ordered sum -> deterministic)
// ---------------------------------------------------------------------------
__global__ __launch_bounds__(256)
void reduce_partials(const float* __restrict__ P, const float* __restrict__ bias,
                     float* __restrict__ Y, int Nout, int S,
                     const int* __restrict__ hz)
{
    const int q = blockIdx.x * blockDim.x + threadIdx.x;      // float4 index
    if (q >= N_NODES * Nout / 4) return;
    const int g = q * 4;
    const int m = g / Nout;
    const int n = g % Nout;
    float4 acc = {0.0f, 0.0f, 0.0f, 0.0f};
    for (int s = 0; s < S; ++s) {
        const float4 p = *(const float4*)(P + ((size_t)s * N_NODES + m) * Nout + n);
        acc.x += p.x; acc.y += p.y; acc.z += p.z; acc.w += p.w;
    }
    float scale = 1.0f;
    if (hz) {
        float v = (float)hz[0] * 0.001f;
        scale = fminf(fmaxf(v, 0.1f), 1.0f);
    }
    const float4 b = *(const float4*)(bias + n);
    float4 r;
    r.x = (acc.x + b.x) * scale;
    r.y = (acc.y + b.y) * scale;
    r.z = (acc.z + b.z) * scale;
    r.w = (acc.w + b.w) * scale;
    *(float4*)(Y + (size_t)m * Nout + n) = r;
}

// ---------------------------------------------------------------------------
// Copy 27 valid rows into a 32-row buffer, zero-filling rows 27..31.
// ---------------------------------------------------------------------------
__global__ __launch_bounds__(256)
void pad_rows(const float* __restrict__ src, float* __restrict__ dst, int K)
{
    const int q = blockIdx.x * blockDim.x + threadIdx.x;      // float4 index
    if (q >= MPAD * K / 4) return;
    const int g = q * 4;
    float4 v = {0.0f, 0.0f, 0.0f, 0.0f};
    if (g / K < N_NODES) v = *(const float4*)(src + g);
    *(float4*)(dst + g) = v;
}

__global__ __launch_bounds__(256)
void zero_fill(float* __restrict__ p, int n4)
{
    const int q = blockIdx.x * blockDim.x + threadIdx.x;
    if (q < n4) {
        const float4 z = {0.0f, 0.0f, 0.0f, 0.0f};
        *(float4*)(p + (size_t)q * 4) = z;
    }
}

// ---------------------------------------------------------------------------
// Granger dots: col[i] = node[i,:].w_n ; row[i] = hist[i,:].w_h
// ---------------------------------------------------------------------------
__global__ __launch_bounds__(256)
void granger_dot(const float* __restrict__ node, const float* __restrict__ hist,
                 const float* __restrict__ gw,
                 float* __restrict__ col, float* __restrict__ row)
{
    __shared__ float sc[256], sr[256];
    const int i = blockIdx.x;
    const int t = threadIdx.x;
    const float4* np  = (const float4*)(node + (size_t)i * DK);
    const float4* hp  = (const float4*)(hist + (size_t)i * DK);
    const float4* wnp = (const float4*)(gw + DK);    // w_n = granger_w[E:]
    const float4* whp = (const float4*)(gw);         // w_h = granger_w[:E]
    float accC = 0.0f, accR = 0.0f;
    for (int c = t; c < DK / 4; c += 256) {
        float4 a = np[c], w1 = wnp[c];
        float4 b = hp[c], w2 = whp[c];
        accC += a.x * w1.x + a.y * w1.y + a.z * w1.z + a.w * w1.w;
        accR += b.x * w2.x + b.y * w2.y + b.z * w2.z + b.w * w2.w;
    }
    sc[t] = accC; sr[t] = accR;
    __syncthreads();
    for (int s = 128; s > 0; s >>= 1) {
        if (t < s) { sc[t] += sc[t + s]; sr[t] += sr[t + s]; }
        __syncthreads();
    }
    if (t == 0) { col[i] = sc[0]; row[i] = sr[0]; }
}

// ---------------------------------------------------------------------------
// adj[i,j] = sigmoid(col[i] + row[j] + b), zeroed diagonal
// ---------------------------------------------------------------------------
__global__ void adj_kernel(const float* __restrict__ col, const float* __restrict__ row,
                           const float* __restrict__ gb, float* __restrict__ outAdj)
{
    const int t = blockIdx.x * blockDim.x + threadIdx.x;
    if (t >= N_NODES * N_NODES) return;
    const int i = t / N_NODES, j = t % N_NODES;
    const float x = col[i] + row[j] + gb[0];
    const float a = 1.0f / (1.0f + __expf(-x));
    outAdj[t] = (i == j) ? 0.0f : a;
}

// ---------------------------------------------------------------------------
// One block per head: scores (27x27), softmax, O = attn @ V (27x1280)
// ---------------------------------------------------------------------------
__global__ __launch_bounds__(256)
void attention_kernel(const float* __restrict__ qkv, float* __restrict__ o)
{
    __shared__ float sS[N_NODES][N_NODES + 1];
    const int h = blockIdx.x;
    const int t = threadIdx.x;
    const size_t qoff = (size_t)h * 1280;
    const size_t koff = 5120 + (size_t)h * 1280;
    const size_t voff = 10240 + (size_t)h * 1280;
    const float rsc = 0.027950849718747373f;     // 1/sqrt(1280)

    for (int p = t; p < N_NODES * N_NODES; p += 256) {
        const int i = p / N_NODES, j = p % N_NODES;
        const float4* q  = (const float4*)(qkv + (size_t)i * E3 + qoff);
        const float4* kk = (const float4*)(qkv + (size_t)j * E3 + koff);
        float acc = 0.0f;
        for (int c = 0; c < 1280 / 4; ++c) {
            float4 a = q[c], b = kk[c];
            acc += a.x * b.x + a.y * b.y + a.z * b.z + a.w * b.w;
        }
        sS[i][j] = acc * rsc;
    }
    __syncthreads();

    if (t < N_NODES) {
        float mx = -1e30f;
        for (int j = 0; j < N_NODES; ++j) mx = fmaxf(mx, sS[t][j]);
        float sum = 0.0f;
        for (int j = 0; j < N_NODES; ++j) {
            const float e = __expf(sS[t][j] - mx);
            sS[t][j] = e; sum += e;
        }
        const float inv = 1.0f / sum;
        for (int j = 0; j < N_NODES; ++j) sS[t][j] *= inv;
    }
    __syncthreads();

    for (int p = t; p < N_NODES * (1280 / 4); p += 256) {
        const int i  = p / 320;
        const int dq = p % 320;
        float4 acc = {0.0f, 0.0f, 0.0f, 0.0f};
        for (int j = 0; j < N_NODES; ++j) {
            const float a = sS[i][j];
            const float4 vv = *(const float4*)(qkv + (size_t)j * E3 + voff + 4 * dq);
            acc.x += a * vv.x; acc.y += a * vv.y; acc.z += a * vv.z; acc.w += a * vv.w;
        }
        *(float4*)(o + (size_t)i * DK + qoff + 4 * dq) = acc;
    }
}

extern "C" void kernel_launch(void* const* d_in, const int* in_sizes, int n_in,
                              void* d_out, int out_size, void* d_ws, size_t ws_size,
                              hipStream_t stream) {
    const float* node = (const float*)d_in[0];
    const float* hist = (const float*)d_in[1];
    const float* gw   = (const float*)d_in[2];
    const float* gb   = (const float*)d_in[3];
    const float* inW  = (const float*)d_in[4];
    const float* inB  = (const float*)d_in[5];
    const float* outW = (const float*)d_in[6];
    const float* outB = (const float*)d_in[7];
    const int*   hz   = (const int*)d_in[8];

    float* out = (float*)d_out;                        // [27*5120]
    float* adj = out + (size_t)N_NODES * DK;           // [729]

    // Workspace layout (floats)
    float* qkv     = (float*)d_ws;                     // 27*15360 = 414720
    float* oPad    = qkv  + (size_t)N_NODES * E3;      // 32*5120  = 163840
    float* xPad    = oPad + (size_t)MPAD * DK;         // 32*5120  = 163840
    float* col     = xPad + (size_t)MPAD * DK;         // 27
    float* row     = col + 32;                         // 27 (padded to 32)
    float* P       = row + 32;                         // split-K partials (shared)
    const size_t baseFloats = (size_t)N_NODES * E3 + 2 * (size_t)MPAD * DK + 64;

    // Pick split-K factor S (power of two, <=16) that fits in workspace.
    // Partial buffer is reused by both GEMMs; GEMM1's (27*15360 per split) is larger.
    const size_t availFloats = (ws_size / 4 > baseFloats) ? (ws_size / 4 - baseFloats) : 0;
    int S = 1;
    while (S < 16 && (size_t)(2 * S) * (size_t)N_NODES * E3 <= availFloats) S *= 2;
    const int Kc = DK / S;                             // 5120,2560,...,320 (all %32==0)

    // Stage padded activations; zero rows 27..31 of the attention output buffer.
    pad_rows<<<(MPAD * DK / 4 + 255) / 256, 256, 0, stream>>>(node, xPad, DK);
    zero_fill<<<((MPAD - N_NODES) * DK / 4 + 255) / 256, 256, 0, stream>>>(
        oPad + (size_t)N_NODES * DK, (MPAD - N_NODES) * DK / 4);

    granger_dot<<<N_NODES, 256, 0, stream>>>(node, hist, gw, col, row);
    adj_kernel<<<3, 256, 0, stream>>>(col, row, gb, adj);

    // qkv = X @ in_proj_w^T + in_proj_b   (480 N-tiles x S K-chunks)
    gemm_xwt_splitk<<<dim3(E3 / 128, S), 128, 0, stream>>>(xPad, inW, P, DK, Kc, E3);
    reduce_partials<<<(N_NODES * E3 / 4 + 255) / 256, 256, 0, stream>>>(
        P, inB, qkv, E3, S, nullptr);

    attention_kernel<<<4, 256, 0, stream>>>(qkv, oPad);

    // out = (O @ out_proj_w^T + out_proj_b) * hertz   (160 N-tiles x S K-chunks)
    gemm_xwt_splitk<<<dim3(DK / 128, S), 128, 0, stream>>>(oPad, outW, P, DK, Kc, DK);
    reduce_partials<<<(N_NODES * DK / 4 + 255) / 256, 256, 0, stream>>>(
        P, outB, out, DK, S, hz);
}